// SamplingClassifier_44195213476038
// MI455X (gfx1250) — compile-verified
//
#include <hip/hip_runtime.h>
#include <math.h>

typedef __attribute__((ext_vector_type(2))) float v2f;
typedef __attribute__((ext_vector_type(8))) float v8f;

#define NROWS 4096      // B*T
#define DIM   64
#define KNEG  512
#define NLOG  (KNEG + 1)  // 513
#define NCLS  1000000

// ---------------------------------------------------------------------------
// Kernel 1: x = embeds(4096x64) @ proj_w(64x64) + proj_b, via f32 WMMA.
// Block = 128 threads (4 waves). Wave w computes the 16x16 output tile at
// (M = blockIdx.x*16, N = w*16), chaining 16 x V_WMMA_F32_16X16X4_F32 (K=64).
// ---------------------------------------------------------------------------
__global__ __launch_bounds__(128) void proj_wmma_kernel(
    const float* __restrict__ embeds,   // [4096, 64] row-major
    const float* __restrict__ W,        // [64, 64]   row-major (d_in x d_out)
    const float* __restrict__ bias,     // [64]
    float* __restrict__ x)              // [4096, 64] out
{
    const int wave = threadIdx.x >> 5;        // 0..3 -> N tile
    const int lane = threadIdx.x & 31;
    const int half = lane >> 4;               // 0 | 1
    const int m    = (blockIdx.x << 4) + (lane & 15);   // A-row for this lane
    const int n    = (wave << 4) + (lane & 15);         // B/D column
    const int kh   = half << 1;               // 0 or 2 within each K=4 chunk

    const float* __restrict__ arow = embeds + (size_t)m * DIM;

    v8f c = {};
#pragma unroll
    for (int kb = 0; kb < 16; ++kb) {
        const int k0 = (kb << 2) + kh;
        v2f a, b;
        a[0] = arow[k0];
        a[1] = arow[k0 + 1];
        b[0] = W[(size_t)k0 * DIM + n];
        b[1] = W[(size_t)(k0 + 1) * DIM + n];
        // 8 args: (neg_a, A, neg_b, B, c_mod, C, reuse_a, reuse_b)
        c = __builtin_amdgcn_wmma_f32_16x16x4_f32(
                false, a, false, b, (short)0, c, false, false);
    }

    const float bn = bias[n];
    // C/D layout: VGPR v -> row (v + 8*half), col = lane&15 (+ wave offset in n)
    const int mbase = (blockIdx.x << 4) + (half << 3);
#pragma unroll
    for (int v = 0; v < 8; ++v) {
        x[(size_t)(mbase + v) * DIM + n] = c[v] + bn;
    }
}

// ---------------------------------------------------------------------------
// Kernel 2: per-row scores + logsumexp partial. One block (256 thr) per row n.
//   j = 0      -> pos score  (emb_table[labels[n]] . x[n])
//   j = 1..512 -> neg scores (emb_table[neg[n][j-1]] . x[n])
// Writes logits[n, 0..512] and partials[n] = logsumexp - pos_score.
// Bandwidth-bound: each score streams a contiguous 256B table row (float4).
// ---------------------------------------------------------------------------
__global__ __launch_bounds__(256) void scores_kernel(
    const float* __restrict__ x,        // [4096, 64]
    const int*   __restrict__ labels,   // [4096]
    const int*   __restrict__ negs,     // [4096, 512]
    const float* __restrict__ table,    // [1e6, 64]
    float* __restrict__ logits,         // [4096, 513]
    float* __restrict__ partials)       // [4096]
{
    const int n = blockIdx.x;
    const int t = threadIdx.x;

    __shared__ __align__(16) float xs[DIM];
    __shared__ float red[256];
    __shared__ float pos_s;

    if (t < DIM) xs[t] = x[(size_t)n * DIM + t];
    __syncthreads();
    const float4* __restrict__ x4 = (const float4*)xs;

    // Gather row indices for this thread (<=3), clamp defensively, prefetch.
    int   rows[3];
    int   cnt = 0;
#pragma unroll
    for (int it = 0; it < 3; ++it) {
        const int j = t + it * 256;
        if (j < NLOG) {
            int r = (j == 0) ? labels[n] : negs[(size_t)n * KNEG + (j - 1)];
            r = (r < 0) ? 0 : ((r >= NCLS) ? NCLS - 1 : r);
            rows[it] = r;
            __builtin_prefetch(table + (size_t)r * DIM, 0, 1);  // global_prefetch_b8
            ++cnt;
        }
    }

    float scores[3];
    float tmax = -INFINITY;
    for (int it = 0; it < cnt; ++it) {
        const int j = t + it * 256;
        const float4* __restrict__ e4 = (const float4*)(table + (size_t)rows[it] * DIM);
        float acc = 0.f;
#pragma unroll
        for (int q = 0; q < DIM / 4; ++q) {
            const float4 ev = e4[q];
            const float4 xv = x4[q];
            acc = fmaf(ev.x, xv.x, acc);
            acc = fmaf(ev.y, xv.y, acc);
            acc = fmaf(ev.z, xv.z, acc);
            acc = fmaf(ev.w, xv.w, acc);
        }
        scores[it] = acc;
        tmax = fmaxf(tmax, acc);
        logits[(size_t)n * NLOG + j] = acc;   // coalesced across threads
        if (j == 0) pos_s = acc;              // only thread 0, it==0
    }

    // Block max (logsumexp stabilization)
    red[t] = tmax;
    __syncthreads();
    for (int s = 128; s > 0; s >>= 1) {
        if (t < s) red[t] = fmaxf(red[t], red[t + s]);
        __syncthreads();
    }
    const float bmax = red[0];
    __syncthreads();

    // Block sum of exp
    float esum = 0.f;
    for (int it = 0; it < cnt; ++it) esum += __expf(scores[it] - bmax);
    red[t] = esum;
    __syncthreads();
    for (int s = 128; s > 0; s >>= 1) {
        if (t < s) red[t] += red[t + s];
        __syncthreads();
    }

    if (t == 0) {
        const float lse = bmax + __logf(red[0]);
        partials[n] = lse - pos_s;            // TEMPERATURE == 1.0
    }
}

// ---------------------------------------------------------------------------
// Kernel 3: deterministic tree-reduce of the 4096 partials -> mean loss.
// ---------------------------------------------------------------------------
__global__ __launch_bounds__(256) void loss_reduce_kernel(
    const float* __restrict__ partials, float* __restrict__ loss_out)
{
    __shared__ float red[256];
    const int t = threadIdx.x;
    float s = 0.f;
    for (int i = t; i < NROWS; i += 256) s += partials[i];
    red[t] = s;
    __syncthreads();
    for (int k = 128; k > 0; k >>= 1) {
        if (t < k) red[t] += red[t + k];
        __syncthreads();
    }
    if (t == 0) loss_out[0] = red[0] * (1.0f / (float)NROWS);
}

// ---------------------------------------------------------------------------
extern "C" void kernel_launch(void* const* d_in, const int* in_sizes, int n_in,
                              void* d_out, int out_size, void* d_ws, size_t ws_size,
                              hipStream_t stream)
{
    const float* embeds = (const float*)d_in[0];   // [32,128,64] f32
    const int*   labels = (const int*)  d_in[1];   // [32,128]
    const int*   negs   = (const int*)  d_in[2];   // [4096,512]
    const float* W      = (const float*)d_in[3];   // [64,64]
    const float* bias   = (const float*)d_in[4];   // [64]
    const float* table  = (const float*)d_in[5];   // [1e6,64]

    float* logits   = (float*)d_out;                        // [4096,513]
    float* loss     = logits + (size_t)NROWS * NLOG;        // scalar
    float* x        = (float*)d_ws;                         // [4096,64]
    float* partials = x + (size_t)NROWS * DIM;              // [4096]

    proj_wmma_kernel<<<NROWS / 16, 128, 0, stream>>>(embeds, W, bias, x);
    scores_kernel<<<NROWS, 256, 0, stream>>>(x, labels, negs, table, logits, partials);
    loss_reduce_kernel<<<1, 256, 0, stream>>>(partials, loss);
}